// ModalGNN_56435870269776
// MI455X (gfx1250) — compile-verified
//
#include <hip/hip_runtime.h>
#include <math.h>

typedef __attribute__((ext_vector_type(2))) float v2f;
typedef __attribute__((ext_vector_type(8))) float v8f;

#define D_IN   256
#define D_HID  256
#define D_OUT  128
#define NEG_SLOPE 0.2f
#define EPS_SM 1e-16f
#define EPS_LN 1e-5f

#define BLK_ROWS 128   // 8 waves x 16 rows
#define KC       64    // K chunk staged in LDS
#define LDA      68    // padded A row stride in words (conflict-free: bank = 4r+{0..3})

// ---- monotonic float <-> uint encoding for atomic segment-max ----
__device__ __forceinline__ unsigned enc_f(float f) {
  unsigned u = __float_as_uint(f);
  return (u & 0x80000000u) ? ~u : (u | 0x80000000u);
}
__device__ __forceinline__ float dec_f(unsigned u) {
  return (u & 0x80000000u) ? __uint_as_float(u ^ 0x80000000u)
                           : __uint_as_float(~u);
}

// ------------------------------------------------------------------
// fp32 WMMA GEMM: C[M,N] = A[M,K] @ B[K,N] (+bias, optional exact GELU)
// Block: 256 threads = 8 waves -> 128 rows x 16 cols of C.
// K chunked by 64 through LDS; V_WMMA_F32_16X16X4_F32, K stepped by 4.
// K, N compile-time -> all global/LDS offsets fold to immediates.
// A frag (16x4): lanes 0-15 -> K={k0,k0+1}, lanes 16-31 -> K={k0+2,k0+3}
// B frag (4x16): same K split, N = lane&15
// C/D: VGPR i, lanes 0-15 -> M=i, lanes 16-31 -> M=i+8, N = lane&15
// ------------------------------------------------------------------
template <int K, int N, bool GELU, bool BIAS>
__global__ void wmma_gemm(const float* __restrict__ A,
                          const float* __restrict__ B,
                          const float* __restrict__ bias,
                          float* __restrict__ C, int M) {
  __shared__ float As[BLK_ROWS * LDA];  // 34816 B
  __shared__ float Bs[KC * 16];         //  4096 B

  const int tid     = threadIdx.x;
  const int lane    = tid & 31;
  const int wave    = tid >> 5;
  const int rowBase = blockIdx.y * BLK_ROWS;
  const int colTile = blockIdx.x * 16;
  const int r       = lane & 15;
  const int ko      = (lane >> 4) << 1;          // 0 or 2
  const int waveRow = wave * 16;                 // wave's local row tile
  const bool active = (rowBase + waveRow + 16) <= M;  // wave-uniform

  v8f acc = {};
  for (int kc = 0; kc < K; kc += KC) {
    // ---- stage A chunk: BLK_ROWS x KC (coalesced float4, zero-pad OOB rows)
#pragma unroll
    for (int i = 0; i < 8; ++i) {
      int j    = tid + i * 256;  // 0..2047
      int row  = j >> 4;
      int c4   = j & 15;
      int grow = rowBase + row;
      float4 v = make_float4(0.f, 0.f, 0.f, 0.f);
      if (grow < M) v = *(const float4*)(A + (size_t)grow * K + kc + c4 * 4);
      *(float4*)(&As[row * LDA + c4 * 4]) = v;
    }
    // ---- stage B chunk: KC x 16 (shared by all 8 waves)
    {
      int kk = tid >> 2;  // 0..63
      int c4 = tid & 3;
      float4 v = *(const float4*)(B + (size_t)(kc + kk) * N + colTile + c4 * 4);
      *(float4*)(&Bs[kk * 16 + c4 * 4]) = v;
    }
    __syncthreads();

    if (active) {
      const float* ap = &As[(waveRow + r) * LDA + ko];
      const float* bp = &Bs[ko * 16 + r];
#pragma unroll
      for (int k0 = 0; k0 < KC; k0 += 4) {
        v2f a, b;
        a.x = ap[k0];
        a.y = ap[k0 + 1];
        b.x = bp[k0 * 16];
        b.y = bp[(k0 + 1) * 16];
        acc = __builtin_amdgcn_wmma_f32_16x16x4_f32(false, a, false, b,
                                                    (short)0, acc, false, false);
      }
    }
    __syncthreads();
  }
  if (!active) return;

#pragma unroll
  for (int i = 0; i < 8; ++i) {
    int row = rowBase + waveRow + ((lane < 16) ? i : (i + 8));
    int col = colTile + r;
    float v = acc[i];
    if (BIAS) v += bias[col];
    if (GELU) v = 0.5f * v * (1.0f + erff(v * 0.70710678118654752f));
    C[(size_t)row * N + col] = v;
  }
}

// one wave per node: a_src[i] = g[i,:] . att_src ; a_dst likewise (D_OUT=128)
__global__ void att_scores_kernel(const float* __restrict__ g,
                                  const float* __restrict__ att_src,
                                  const float* __restrict__ att_dst,
                                  float* __restrict__ a_src,
                                  float* __restrict__ a_dst, int N) {
  int node = blockIdx.x * (blockDim.x >> 5) + (threadIdx.x >> 5);
  int lane = threadIdx.x & 31;
  if (node >= N) return;
  float4 gv = ((const float4*)(g + (size_t)node * D_OUT))[lane];
  float4 as = ((const float4*)att_src)[lane];
  float4 ad = ((const float4*)att_dst)[lane];
  float s = gv.x * as.x + gv.y * as.y + gv.z * as.z + gv.w * as.w;
  float d = gv.x * ad.x + gv.y * ad.y + gv.z * ad.z + gv.w * ad.w;
  for (int off = 16; off > 0; off >>= 1) {
    s += __shfl_xor(s, off, 32);
    d += __shfl_xor(d, off, 32);
  }
  if (lane == 0) { a_src[node] = s; a_dst[node] = d; }
}

__global__ void init_buffers_kernel(float* __restrict__ out,
                                    float* __restrict__ denom,
                                    unsigned* __restrict__ emax, int N) {
  int i = blockIdx.x * blockDim.x + threadIdx.x;
  if (i < N * D_OUT) out[i] = 0.0f;
  if (i < N) { denom[i] = 0.0f; emax[i] = 0u; }  // 0 < enc(-inf)=0x007FFFFF
}

// edge_index layout: ei[0..E-1]=src row, ei[E..2E-1]=dst row; e>=E => self-loop
__device__ __forceinline__ void edge_nodes(const int* __restrict__ ei, int e,
                                           int E, int* s, int* d) {
  if (e < E) { *s = ei[e]; *d = ei[E + e]; }
  else       { *s = *d = e - E; }
}

__device__ __forceinline__ float edge_score(const float* __restrict__ a_src,
                                            const float* __restrict__ a_dst,
                                            int s, int d) {
  float v = a_src[s] + a_dst[d];
  return (v > 0.0f) ? v : NEG_SLOPE * v;
}

__global__ void edge_max_kernel(const int* __restrict__ ei, int E, int N,
                                const float* __restrict__ a_src,
                                const float* __restrict__ a_dst,
                                unsigned* __restrict__ emax) {
  int e = blockIdx.x * blockDim.x + threadIdx.x;
  if (e >= E + N) return;
  int s, d; edge_nodes(ei, e, E, &s, &d);
  atomicMax(&emax[d], enc_f(edge_score(a_src, a_dst, s, d)));
}

__global__ void edge_sum_kernel(const int* __restrict__ ei, int E, int N,
                                const float* __restrict__ a_src,
                                const float* __restrict__ a_dst,
                                const unsigned* __restrict__ emax,
                                float* __restrict__ denom) {
  int e = blockIdx.x * blockDim.x + threadIdx.x;
  if (e >= E + N) return;
  int s, d; edge_nodes(ei, e, E, &s, &d);
  float p = expf(edge_score(a_src, a_dst, s, d) - dec_f(emax[d]));
  atomicAdd(&denom[d], p);
}

// one wave per edge: out[dst,:] += alpha * g[src,:]  (128 = 32 lanes x float4)
__global__ void edge_scatter_kernel(const int* __restrict__ ei, int E, int N,
                                    const float* __restrict__ a_src,
                                    const float* __restrict__ a_dst,
                                    const unsigned* __restrict__ emax,
                                    const float* __restrict__ denom,
                                    const float* __restrict__ g,
                                    float* __restrict__ out) {
  int w    = blockIdx.x * (blockDim.x >> 5) + (threadIdx.x >> 5);
  int lane = threadIdx.x & 31;
  if (w >= E + N) return;
  int s, d; edge_nodes(ei, w, E, &s, &d);
  float v = edge_score(a_src, a_dst, s, d);
  float alpha = expf(v - dec_f(emax[d])) / (denom[d] + EPS_SM);
  float4 gv = ((const float4*)(g + (size_t)s * D_OUT))[lane];
  float* op = out + (size_t)d * D_OUT + lane * 4;
  atomicAdd(op + 0, alpha * gv.x);
  atomicAdd(op + 1, alpha * gv.y);
  atomicAdd(op + 2, alpha * gv.z);
  atomicAdd(op + 3, alpha * gv.w);
}

// one wave per node: out = LN(out + b_gat) * gamma + beta
__global__ void layernorm_kernel(float* __restrict__ out,
                                 const float* __restrict__ b_gat,
                                 const float* __restrict__ gamma,
                                 const float* __restrict__ beta, int N) {
  int node = blockIdx.x * (blockDim.x >> 5) + (threadIdx.x >> 5);
  int lane = threadIdx.x & 31;
  if (node >= N) return;
  float4* row = (float4*)(out + (size_t)node * D_OUT);
  float4 v = row[lane];
  float4 bb = ((const float4*)b_gat)[lane];
  v.x += bb.x; v.y += bb.y; v.z += bb.z; v.w += bb.w;
  float sum = v.x + v.y + v.z + v.w;
  for (int off = 16; off > 0; off >>= 1) sum += __shfl_xor(sum, off, 32);
  float mu = sum * (1.0f / D_OUT);
  float dx = v.x - mu, dy = v.y - mu, dz = v.z - mu, dw = v.w - mu;
  float q = dx * dx + dy * dy + dz * dz + dw * dw;
  for (int off = 16; off > 0; off >>= 1) q += __shfl_xor(q, off, 32);
  float inv = rsqrtf(q * (1.0f / D_OUT) + EPS_LN);
  float4 gm = ((const float4*)gamma)[lane];
  float4 bt = ((const float4*)beta)[lane];
  float4 o;
  o.x = dx * inv * gm.x + bt.x;
  o.y = dy * inv * gm.y + bt.y;
  o.z = dz * inv * gm.z + bt.z;
  o.w = dw * inv * gm.w + bt.w;
  row[lane] = o;
}

extern "C" void kernel_launch(void* const* d_in, const int* in_sizes, int n_in,
                              void* d_out, int out_size, void* d_ws, size_t ws_size,
                              hipStream_t stream) {
  const float* x        = (const float*)d_in[0];
  const int*   ei       = (const int*)d_in[1];   // int32 (JAX default x64-off)
  const float* W_proj   = (const float*)d_in[2];
  const float* b_proj   = (const float*)d_in[3];
  const float* W_gat    = (const float*)d_in[4];
  const float* att_src  = (const float*)d_in[5];
  const float* att_dst  = (const float*)d_in[6];
  const float* b_gat    = (const float*)d_in[7];
  const float* gamma    = (const float*)d_in[8];
  const float* beta     = (const float*)d_in[9];

  const int N = in_sizes[0] / D_IN;   // 50000
  const int E = in_sizes[1] / 2;      // 800000
  const int Et = E + N;               // with self-loops
  float* out = (float*)d_out;

  // workspace layout (fp32 units)
  float*    h     = (float*)d_ws;                    // [N, 256]
  float*    g     = h + (size_t)N * D_HID;           // [N, 128]
  float*    a_src = g + (size_t)N * D_OUT;           // [N]
  float*    a_dst = a_src + N;                       // [N]
  float*    denom = a_dst + N;                       // [N]
  unsigned* emax  = (unsigned*)(denom + N);          // [N]

  dim3 blk(256);
  const int rowBlks = (N + BLK_ROWS - 1) / BLK_ROWS; // 391

  // 1) h = GELU(x @ W_proj + b_proj)
  wmma_gemm<D_IN, D_HID, true, true>
      <<<dim3(D_HID / 16, rowBlks), blk, 0, stream>>>(x, W_proj, b_proj, h, N);

  // 2) g = h @ W_gat
  wmma_gemm<D_HID, D_OUT, false, false>
      <<<dim3(D_OUT / 16, rowBlks), blk, 0, stream>>>(h, W_gat, nullptr, g, N);

  // 3) per-node attention scores
  att_scores_kernel<<<(N + 7) / 8, blk, 0, stream>>>(g, att_src, att_dst,
                                                     a_src, a_dst, N);

  // 4) init accumulators (re-done every call: graph-replay safe)
  init_buffers_kernel<<<((size_t)N * D_OUT + 255) / 256, blk, 0, stream>>>(
      out, denom, emax, N);

  // 5) segment softmax (max, sum) + weighted scatter
  edge_max_kernel<<<(Et + 255) / 256, blk, 0, stream>>>(ei, E, N, a_src, a_dst, emax);
  edge_sum_kernel<<<(Et + 255) / 256, blk, 0, stream>>>(ei, E, N, a_src, a_dst,
                                                        emax, denom);
  edge_scatter_kernel<<<(Et + 7) / 8, blk, 0, stream>>>(ei, E, N, a_src, a_dst,
                                                        emax, denom, g, out);

  // 6) bias + LayerNorm
  layernorm_kernel<<<(N + 7) / 8, blk, 0, stream>>>(out, b_gat, gamma, beta, N);
}